// GNNPortScoreFaultAware_37306085933445
// MI455X (gfx1250) — compile-verified
//
#include <hip/hip_runtime.h>
#include <math.h>

// ---------------------------------------------------------------------------
// GATv2 x3 for gfx1250 (MI455X).
//  - Dense node transforms: fp32 WMMA (v_wmma_f32_16x16x4_f32), exact 16x16x4
//    tiling (N=50000=3125*16, HC in {64,32}, K in {12,64}).
//  - Edge phase: 16 lanes per edge, float4/float2 row gathers (b128/b64),
//    __shfl_xor head reductions, atomic segment max/sum (tables are L2-resident).
// ---------------------------------------------------------------------------

#define NEG_SLOPE 0.2f
#define EPS_SM    1e-16f
#define EPS_LN    1e-5f

typedef float v2f __attribute__((ext_vector_type(2)));
typedef float v4f __attribute__((ext_vector_type(4)));
typedef float v8f __attribute__((ext_vector_type(8)));

template <int W> struct vecf_sel;
template <> struct vecf_sel<2> { using T = v2f; };
template <> struct vecf_sel<4> { using T = v4f; };

// ------------------------------ utility kernels ----------------------------

__global__ void fill_f32(float* __restrict__ p, float v, long long n) {
  long long i = (long long)blockIdx.x * blockDim.x + threadIdx.x;
  if (i < n) p[i] = v;
}

__global__ void reduce_sum_f32(const float* __restrict__ a, long long n,
                               float* __restrict__ out) {
  long long i = (long long)blockIdx.x * blockDim.x + threadIdx.x;
  long long stride = (long long)gridDim.x * blockDim.x;
  float s = 0.0f;
  for (long long j = i; j < n; j += stride) s += a[j];
  atomicAdd(out, s);
}

__device__ __forceinline__ void atomic_max_f32(float* addr, float val) {
  // valid when *addr is initialized to -inf; monotone-bit-pattern trick
  if (val >= 0.0f)
    atomicMax((int*)addr, __float_as_int(val));
  else
    atomicMin((unsigned int*)addr, (unsigned int)__float_as_int(val));
}

// ---------------- fused dual GEMM: xl = h@Wl+bl, xr = h@Wr+br --------------
// One wave per 16x16 output tile; fp32 WMMA 16x16x4.
// A layout (f32 16x4): lane L holds row L%16, K-pair 2*(L/16) in 2 VGPRs.
// B layout mirrored;  C/D: VGPR i -> row i+8*(L/16), col L%16.
__global__ void __launch_bounds__(256)
gat_gemm_xlxr(const float* __restrict__ h,
              const float* __restrict__ Wl, const float* __restrict__ bl,
              const float* __restrict__ Wr, const float* __restrict__ br,
              float* __restrict__ xl, float* __restrict__ xr,
              int N, int F, int HC)
{
  const int lane   = threadIdx.x & 31;
  const int wave   = blockIdx.x * (blockDim.x >> 5) + (threadIdx.x >> 5);
  const int ntiles = HC >> 4;
  const int mtiles = (N + 15) >> 4;
  if (wave >= mtiles * ntiles) return;

  const int mt   = wave / ntiles;
  const int nt   = wave - mt * ntiles;
  const int row0 = mt << 4;
  const int col0 = nt << 4;
  const int lrow = lane & 15;
  const int kh   = lane >> 4;           // selects K-pair {0,1} vs {2,3}

  int arow = row0 + lrow; if (arow > N - 1) arow = N - 1;   // clamp (keeps EXEC full)
  const float* __restrict__ ap = h + (size_t)arow * F + 2 * kh;
  const int bcol = col0 + lrow;

  v8f cl = {}; v8f cr = {};
  for (int k = 0; k < F; k += 4) {
    v2f a;  a.x = ap[k];  a.y = ap[k + 1];
    const int krow = k + 2 * kh;
    v2f bL; bL.x = Wl[(size_t)krow * HC + bcol]; bL.y = Wl[(size_t)(krow + 1) * HC + bcol];
    v2f bR; bR.x = Wr[(size_t)krow * HC + bcol]; bR.y = Wr[(size_t)(krow + 1) * HC + bcol];
    cl = __builtin_amdgcn_wmma_f32_16x16x4_f32(false, a, false, bL, (short)0, cl, false, false);
    cr = __builtin_amdgcn_wmma_f32_16x16x4_f32(false, a, false, bR, (short)0, cr, false, false);
  }

  const float bbl = bl[bcol];
  const float bbr = br[bcol];
#pragma unroll
  for (int i = 0; i < 8; ++i) {
    const int m = row0 + i + 8 * kh;
    if (m < N) {
      xl[(size_t)m * HC + bcol] = cl[i] + bbl;
      xr[(size_t)m * HC + bcol] = cr[i] + bbr;
    }
  }
}

// ------------------------------ edge kernels -------------------------------
// 16 lanes per edge; lane l owns channels [l*CPL, l*CPL+CPL).
// Edges e >= E are the self loops (e-E, e-E) with edge_attr = mean(edge_attr).

template <int H, int C>
__global__ void __launch_bounds__(256)
edge_logits_v(const float* __restrict__ xl, const float* __restrict__ xr,
              const int* __restrict__ src, const int* __restrict__ dst,
              const float* __restrict__ ea, const float* __restrict__ easum,
              float inv_e,
              const float* __restrict__ We, const float* __restrict__ att,
              float* __restrict__ logits, float* __restrict__ lmax,
              int E, int N)
{
  constexpr int HC  = H * C;
  constexpr int CPL = HC / 16;     // channels per lane (4 or 2)
  constexpr int LPH = 16 / H;      // lanes per head
  using VT = typename vecf_sel<CPL>::T;

  const long long g = ((long long)blockIdx.x * blockDim.x + threadIdx.x) >> 4;
  const int lane16  = threadIdx.x & 15;
  if (g >= (long long)E + N) return;

  int s = 0, d = 0; float a = 0.0f;
  if (lane16 == 0) {
    if (g < E) { s = src[g]; d = dst[g]; a = ea[g]; }
    else       { s = d = (int)(g - E); a = easum[0] * inv_e; }
  }
  s = __shfl(s, 0, 16);
  d = __shfl(d, 0, 16);
  a = __shfl(a, 0, 16);

  const VT vl = *(const VT*)(xl + (size_t)s * HC + lane16 * CPL);
  const VT vr = *(const VT*)(xr + (size_t)d * HC + lane16 * CPL);
  const VT vw = *(const VT*)(We  + lane16 * CPL);
  const VT va = *(const VT*)(att + lane16 * CPL);

  float acc = 0.0f;
#pragma unroll
  for (int j = 0; j < CPL; ++j) {
    float m  = vl[j] + vr[j] + a * vw[j];
    float lr = m > 0.0f ? m : NEG_SLOPE * m;
    acc += lr * va[j];
  }
#pragma unroll
  for (int off = 1; off < LPH; off <<= 1)
    acc += __shfl_xor(acc, off, 16);

  if ((lane16 & (LPH - 1)) == 0) {
    const int hh = lane16 / LPH;
    logits[(size_t)g * H + hh] = acc;
    atomic_max_f32(&lmax[(size_t)d * H + hh], acc);
  }
}

__global__ void edge_exp(float* logits,                      // in: logits, out: exp()
                         const int* __restrict__ dst,
                         const float* __restrict__ lmax, float* __restrict__ denom,
                         int E, int N, int H)
{
  long long t = (long long)blockIdx.x * blockDim.x + threadIdx.x;
  long long total = (long long)(E + N) * H;
  if (t >= total) return;
  const int hh = (int)(t % H);
  const long long e = t / H;
  const int d = (e < E) ? dst[e] : (int)(e - E);
  float ex = __expf(logits[t] - lmax[(size_t)d * H + hh]);
  logits[t] = ex;
  atomicAdd(&denom[(size_t)d * H + hh], ex);
}

template <int H, int C>
__global__ void __launch_bounds__(256)
edge_agg_v(const float* __restrict__ ex,
           const float* __restrict__ xl,
           const int* __restrict__ src, const int* __restrict__ dst,
           const float* __restrict__ denom,
           float* __restrict__ agg,
           int E, int N)
{
  constexpr int HC  = H * C;
  constexpr int CPL = HC / 16;
  constexpr int LPH = 16 / H;
  using VT = typename vecf_sel<CPL>::T;

  const long long g = ((long long)blockIdx.x * blockDim.x + threadIdx.x) >> 4;
  const int lane16  = threadIdx.x & 15;
  if (g >= (long long)E + N) return;

  int s = 0, d = 0;
  if (lane16 == 0) {
    if (g < E) { s = src[g]; d = dst[g]; }
    else       { s = d = (int)(g - E); }
  }
  s = __shfl(s, 0, 16);
  d = __shfl(d, 0, 16);

  const int hh = lane16 / LPH;
  const float alpha = ex[(size_t)g * H + hh] / (denom[(size_t)d * H + hh] + EPS_SM);

  const VT vl = *(const VT*)(xl + (size_t)s * HC + lane16 * CPL);
  float* __restrict__ po = agg + (size_t)d * HC + lane16 * CPL;
#pragma unroll
  for (int j = 0; j < CPL; ++j)
    atomicAdd(&po[j], vl[j] * alpha);
}

// ------------------------- bias + LayerNorm + ELU --------------------------

template <int D>
__global__ void node_finish_t(const float* __restrict__ agg, const float* __restrict__ bias,
                              const float* __restrict__ g, const float* __restrict__ beta,
                              float* __restrict__ out, int N, int do_elu)
{
  int n = blockIdx.x * blockDim.x + threadIdx.x;
  if (n >= N) return;
  const v4f* __restrict__ ap = (const v4f*)(agg + (size_t)n * D);
  const v4f* __restrict__ bp = (const v4f*)bias;

  float v[D];
  float mu = 0.0f;
#pragma unroll
  for (int j = 0; j < D / 4; ++j) {
    v4f t = ap[j];
    v4f b = bp[j];
#pragma unroll
    for (int k = 0; k < 4; ++k) {
      float y = t[k] + b[k];
      v[4 * j + k] = y;
      mu += y;
    }
  }
  mu *= (1.0f / D);
  float var = 0.0f;
#pragma unroll
  for (int i = 0; i < D; ++i) { float t = v[i] - mu; var += t * t; }
  var *= (1.0f / D);
  const float inv = rsqrtf(var + EPS_LN);
#pragma unroll
  for (int i = 0; i < D; ++i) {
    float y = (v[i] - mu) * inv * g[i] + beta[i];
    if (do_elu) y = y > 0.0f ? y : expm1f(y);
    out[(size_t)n * D + i] = y;
  }
}

// ------------------------------ host driver --------------------------------

static inline unsigned int nblk(long long n, int b) {
  return (unsigned int)((n + b - 1) / b);
}

template <int H, int C>
static void run_gat_layer(hipStream_t stream,
    const float* hin, int F,
    const float* Wl, const float* bl, const float* Wr, const float* br,
    const float* We, const float* att, const float* bparam,
    const float* g, const float* beta,
    const int* src, const int* dst, const float* ea, const float* easum,
    float* xl, float* xr, float* agg, float* logits, float* lmax, float* denom,
    float* hout, int do_elu, int N, int E)
{
  constexpr int HC = H * C;
  const long long NH   = (long long)N * H;
  const long long NHC  = (long long)N * HC;
  const long long GRP  = (long long)E + N;          // edges incl. self loops
  const long long TOT  = GRP * H;                   // (edge, head) pairs

  fill_f32<<<nblk(NH, 256), 256, 0, stream>>>(lmax, -INFINITY, NH);
  fill_f32<<<nblk(NH, 256), 256, 0, stream>>>(denom, 0.0f, NH);
  fill_f32<<<nblk(NHC, 256), 256, 0, stream>>>(agg, 0.0f, NHC);

  const long long waves = (long long)((N + 15) / 16) * (HC / 16);
  gat_gemm_xlxr<<<nblk(waves * 32, 256), 256, 0, stream>>>(
      hin, Wl, bl, Wr, br, xl, xr, N, F, HC);

  const float inv_e = 1.0f / (float)E;
  edge_logits_v<H, C><<<nblk(GRP * 16, 256), 256, 0, stream>>>(
      xl, xr, src, dst, ea, easum, inv_e, We, att, logits, lmax, E, N);
  edge_exp<<<nblk(TOT, 256), 256, 0, stream>>>(logits, dst, lmax, denom, E, N, H);
  edge_agg_v<H, C><<<nblk(GRP * 16, 256), 256, 0, stream>>>(
      logits, xl, src, dst, denom, agg, E, N);

  node_finish_t<HC><<<nblk(N, 256), 256, 0, stream>>>(agg, bparam, g, beta, hout, N, do_elu);
}

extern "C" void kernel_launch(void* const* d_in, const int* in_sizes, int n_in,
                              void* d_out, int out_size, void* d_ws, size_t ws_size,
                              hipStream_t stream)
{
  // setup_inputs() flattened in insertion order:
  // 0:x 1:edge_attr 2..8:p1{Wl,bl,Wr,br,We,att,b} 9..15:p2 16..22:p3
  // 23,24:ln1{g,b} 25,26:ln2 27,28:ln3 29:edge_index
  const float* x  = (const float*)d_in[0];
  const float* ea = (const float*)d_in[1];

  const float* p1[7]; for (int i = 0; i < 7; ++i) p1[i] = (const float*)d_in[2 + i];
  const float* p2[7]; for (int i = 0; i < 7; ++i) p2[i] = (const float*)d_in[9 + i];
  const float* p3[7]; for (int i = 0; i < 7; ++i) p3[i] = (const float*)d_in[16 + i];
  const float* ln1g = (const float*)d_in[23]; const float* ln1b = (const float*)d_in[24];
  const float* ln2g = (const float*)d_in[25]; const float* ln2b = (const float*)d_in[26];
  const float* ln3g = (const float*)d_in[27]; const float* ln3b = (const float*)d_in[28];
  const int*   ei   = (const int*)d_in[29];

  const int N = in_sizes[0] / 12;
  const int E = in_sizes[1];
  const int* src = ei;
  const int* dst = ei + E;

  // ---- workspace layout (floats) ----
  float* ws = (float*)d_ws;
  const size_t NN = (size_t)N;
  const size_t ET = (size_t)E + NN;
  float* xl     = ws; ws += NN * 64;
  float* xr     = ws; ws += NN * 64;
  float* agg    = ws; ws += NN * 64;
  float* h1     = ws; ws += NN * 64;
  float* h2     = ws; ws += NN * 64;
  float* logits = ws; ws += ET * 4;
  float* lmax   = ws; ws += NN * 4;
  float* denom  = ws; ws += NN * 4;
  float* easum  = ws; ws += 1;

  // mean(edge_attr) for self-loop fill value
  fill_f32<<<1, 32, 0, stream>>>(easum, 0.0f, 1);
  reduce_sum_f32<<<256, 256, 0, stream>>>(ea, (long long)E, easum);

  // layer 1: 12 -> 4x16, ELU(LN(.))
  run_gat_layer<4, 16>(stream, x, 12,
                p1[0], p1[1], p1[2], p1[3], p1[4], p1[5], p1[6],
                ln1g, ln1b, src, dst, ea, easum,
                xl, xr, agg, logits, lmax, denom, h1, 1, N, E);
  // layer 2: 64 -> 4x16, ELU(LN(.))
  run_gat_layer<4, 16>(stream, h1, 64,
                p2[0], p2[1], p2[2], p2[3], p2[4], p2[5], p2[6],
                ln2g, ln2b, src, dst, ea, easum,
                xl, xr, agg, logits, lmax, denom, h2, 1, N, E);
  // layer 3: 64 -> 1x32, LN only -> d_out
  run_gat_layer<1, 32>(stream, h2, 64,
                p3[0], p3[1], p3[2], p3[3], p3[4], p3[5], p3[6],
                ln3g, ln3b, src, dst, ea, easum,
                xl, xr, agg, logits, lmax, denom, (float*)d_out, 0, N, E);
}